// LatentStore_66116726555439
// MI455X (gfx1250) — compile-verified
//
#include <hip/hip_runtime.h>

typedef float v2f __attribute__((ext_vector_type(2)));
typedef float v8f __attribute__((ext_vector_type(8)));

constexpr int D      = 256;   // latent size
constexpr int TOPK   = 50;    // max_results (harness scalar; fixed per reference)
constexpr int CHUNK  = 4096;  // rows handled per block in the fused pass
constexpr int TPB    = 256;   // 8 waves (wave32)
constexpr int TILES  = CHUNK / 16;
constexpr int WAVES  = TPB / 32;

// Order-preserving float -> uint mapping (monotone: bigger float -> bigger uint)
__device__ __forceinline__ unsigned mapf(float f) {
  unsigned u = __float_as_uint(f);
  return (u & 0x80000000u) ? ~u : (u | 0x80000000u);
}

// ---------------------------------------------------------------------------
// Pass 1 (fused): WMMA f32 GEMV sims + per-block destructive top-50 in LDS.
// Each wave computes 16 rows' dot products at a time with
// V_WMMA_F32_16X16X4_F32: A = 16x4 tile of latents, B = query chunk
// replicated across all 16 columns, so every column of D accumulates the sims.
// ---------------------------------------------------------------------------
__global__ __launch_bounds__(TPB) void sims_topk_kernel(
    const float* __restrict__ query, const float* __restrict__ latents,
    int n, float* __restrict__ candVal, unsigned* __restrict__ candIdx) {
  __shared__ float qLds[D];
  __shared__ float sims[CHUNK];
  __shared__ unsigned long long red[TPB];

  const int tid        = threadIdx.x;
  const int lane       = tid & 31;
  const int wave       = tid >> 5;
  const int blockStart = blockIdx.x * CHUNK;

  qLds[tid] = query[tid];            // D == TPB == 256
  __syncthreads();

  // A-layout (f32 16x4): lanes 0-15 -> row M=lane, K = c+0,c+1 in VGPR0/1;
  // lanes 16-31 -> row M=lane-16, K = c+2,c+3.  B mirrors this in K.
  const int rowInTile = lane & 15;
  const int kOff      = (lane >> 4) * 2;

  for (int t = wave; t < TILES; t += WAVES) {
    const int gRow0 = blockStart + t * 16;
    if (gRow0 < n) {  // n % 16 == 0 -> tiles are fully valid or fully invalid
      const float* __restrict__ arow =
          latents + (size_t)(gRow0 + rowInTile) * D + kOff;
      v8f acc = {0.f, 0.f, 0.f, 0.f, 0.f, 0.f, 0.f, 0.f};
#pragma unroll 8
      for (int c = 0; c < D; c += 4) {
        float2 af = *(const float2*)(arow + c);
        float2 bf = *(const float2*)(&qLds[c + kOff]);
        v2f a = {af.x, af.y};
        v2f b = {bf.x, bf.y};
        acc = __builtin_amdgcn_wmma_f32_16x16x4_f32(
            false, a, false, b, (short)0, acc, false, false);
      }
      // D-layout: VGPR r -> (lanes 0-15: M=r), (lanes 16-31: M=8+r).
      float* s = &sims[t * 16];
      if (lane == 0) {
        *(float4*)(s + 0) = make_float4(acc[0], acc[1], acc[2], acc[3]);
        *(float4*)(s + 4) = make_float4(acc[4], acc[5], acc[6], acc[7]);
      } else if (lane == 16) {
        *(float4*)(s + 8)  = make_float4(acc[0], acc[1], acc[2], acc[3]);
        *(float4*)(s + 12) = make_float4(acc[4], acc[5], acc[6], acc[7]);
      }
    } else {
      if (lane < 16) sims[t * 16 + lane] = -__builtin_huge_valf();
    }
  }
  __syncthreads();

  // 50 rounds of destructive block-argmax over LDS sims.
  // Key = (mapped sim) << 32 | ~globalIdx  -> max key == max sim, min index.
  for (int r = 0; r < TOPK; ++r) {
    unsigned long long best = 0ull;
    for (int i = tid; i < CHUNK; i += TPB) {
      unsigned long long key =
          ((unsigned long long)mapf(sims[i]) << 32) |
          (unsigned)(0xFFFFFFFFu - (unsigned)(blockStart + i));
      best = (key > best) ? key : best;
    }
    red[tid] = best;
    __syncthreads();
    for (int s = TPB / 2; s > 0; s >>= 1) {
      if (tid < s) { if (red[tid + s] > red[tid]) red[tid] = red[tid + s]; }
      __syncthreads();
    }
    if (tid == 0) {
      unsigned gi = 0xFFFFFFFFu - (unsigned)red[0];
      int li = (int)(gi - (unsigned)blockStart);
      candVal[blockIdx.x * TOPK + r] = sims[li];
      candIdx[blockIdx.x * TOPK + r] = gi;
      sims[li] = -__builtin_huge_valf();
    }
    __syncthreads();
  }
}

// ---------------------------------------------------------------------------
// Pass 2: single block merges all block-local candidates into the global
// top-50 (emitted in descending order, like lax.top_k), then gathers
// metadata rows, sims and the winning latent vectors into d_out.
// d_out layout (flat f32): meta[50,3] @0, sims[50] @150, vals[50,256] @200.
// ---------------------------------------------------------------------------
__global__ __launch_bounds__(TPB) void final_topk_kernel(
    const float* __restrict__ latents, const int* __restrict__ metadata,
    float* __restrict__ candVal, const unsigned* __restrict__ candIdx,
    int nCand, float* __restrict__ out) {
  __shared__ unsigned long long red[TPB];
  __shared__ int   topIdx[TOPK];
  __shared__ float topSim[TOPK];
  const int tid = threadIdx.x;

  for (int r = 0; r < TOPK; ++r) {
    unsigned long long best = 0ull;
    for (int i = tid; i < nCand; i += TPB) {
      unsigned long long key =
          ((unsigned long long)mapf(candVal[i]) << 32) |
          (unsigned)(0xFFFFFFFFu - (unsigned)i);
      best = (key > best) ? key : best;
    }
    red[tid] = best;
    __syncthreads();
    for (int s = TPB / 2; s > 0; s >>= 1) {
      if (tid < s) { if (red[tid + s] > red[tid]) red[tid] = red[tid + s]; }
      __syncthreads();
    }
    if (tid == 0) {
      unsigned i = 0xFFFFFFFFu - (unsigned)red[0];
      topIdx[r] = (int)candIdx[i];
      topSim[r] = candVal[i];
      candVal[i] = -__builtin_huge_valf();   // rewritten by pass 1 each call
    }
    __syncthreads();
  }

  for (int j = tid; j < TOPK * 3; j += TPB) {
    int r = j / 3, c = j % 3;
    out[j] = (float)metadata[(size_t)topIdx[r] * 3 + c];
  }
  if (tid < TOPK) out[TOPK * 3 + tid] = topSim[tid];
  for (int j = tid; j < TOPK * D; j += TPB) {
    int r = j >> 8, c = j & (D - 1);
    out[TOPK * 3 + TOPK + j] = latents[(size_t)topIdx[r] * D + c];
  }
}

extern "C" void kernel_launch(void* const* d_in, const int* in_sizes, int n_in,
                              void* d_out, int out_size, void* d_ws, size_t ws_size,
                              hipStream_t stream) {
  const float* query   = (const float*)d_in[0];
  const float* latents = (const float*)d_in[1];
  const int*   metadata = (const int*)d_in[2];

  const int n = in_sizes[1] / D;                  // 1,000,000
  const int g = (n + CHUNK - 1) / CHUNK;          // 245 blocks

  float*    candVal = (float*)d_ws;
  size_t    valBytes = ((size_t)g * TOPK * sizeof(float) + 255) & ~(size_t)255;
  unsigned* candIdx = (unsigned*)((char*)d_ws + valBytes);

  sims_topk_kernel<<<g, TPB, 0, stream>>>(query, latents, n, candVal, candIdx);
  final_topk_kernel<<<1, TPB, 0, stream>>>(latents, metadata, candVal, candIdx,
                                           g * TOPK, (float*)d_out);
}